// AgentGnn_CRAT_15436112462394
// MI455X (gfx1250) — compile-verified
//
#include <hip/hip_runtime.h>
#include <hip/hip_bf16.h>

typedef float v2f __attribute__((ext_vector_type(2)));
typedef float v8f __attribute__((ext_vector_type(8)));

#define HDIM 128
#define NNODES 16384
#define NSAMP 1024
#define AGENTS 16
#define NOUT 512           // [Pf | Qf | Ps | Qs]
#define ZROW 258           // 2H + D
#define BN_EPS 1e-5f

// ---------------------------------------------------------------------------
// Pack kernel: build Wcat[k][n] (K=128 x N=512, row-major) from wf/ws, plus
// edge-coefficient tables E0/E1 and Bias per output column; zero BN stats.
// Column groups: g=0 Pf (x_i part of wf), g=1 Qf (x_j part), g=2 Ps, g=3 Qs.
// ---------------------------------------------------------------------------
__global__ __launch_bounds__(256) void pack_kernel(
    const float* __restrict__ wf, const float* __restrict__ bf,
    const float* __restrict__ wsm, const float* __restrict__ bs,
    float* __restrict__ Wcat, float* __restrict__ E0, float* __restrict__ E1,
    float* __restrict__ Bias, float* __restrict__ stats) {
  int idx = blockIdx.x * 256 + threadIdx.x;     // 65536 entries
  int k = idx >> 9;
  int n = idx & 511;
  int g = n >> 7;
  int h = n & 127;
  const float* src = (g < 2) ? wf : wsm;
  int koff = (g & 1) ? (HDIM + k) : k;
  Wcat[idx] = src[h * ZROW + koff];
  if (k == 0) {
    float sgn = (g & 1) ? -1.0f : 1.0f;         // +We on target, -We on source
    E0[n] = sgn * src[h * ZROW + 256];
    E1[n] = sgn * src[h * ZROW + 257];
    Bias[n] = (g == 0) ? bf[h] : ((g == 2) ? bs[h] : 0.0f);
  }
  if (idx < 256) stats[idx] = 0.0f;             // sum[128] | sumsq[128]
}

// ---------------------------------------------------------------------------
// Node GEMM: G[N][512] = X[N][128] @ Wcat[128][512] + centers terms + bias.
// One wave computes one 16x16 tile using V_WMMA_F32_16X16X4_F32 (K=4/step).
// A layout (16x4 f32): lane<16 -> M=lane, K={k0,k0+1}; lane>=16 -> K={k0+2,k0+3}.
// B layout (4x16 f32): v0 = row K (lanes0-15) / K+2 (lanes16-31); v1 = K+1/K+3.
// ---------------------------------------------------------------------------
__global__ __launch_bounds__(256) void gemm_pq_kernel(
    const float* __restrict__ X, const float* __restrict__ centers,
    const float* __restrict__ Wcat, const float* __restrict__ E0,
    const float* __restrict__ E1, const float* __restrict__ Bias,
    float* __restrict__ G) {
  int lane = threadIdx.x & 31;
  int wid = threadIdx.x >> 5;
  int tile = blockIdx.x * 8 + wid;              // 32768 tiles total
  int tn = tile & 31;                           // 32 N-tiles
  int tm = tile >> 5;                           // 1024 M-tiles
  int l16 = lane & 15;
  int kh = lane >> 4;                           // K-half select (0/1)

  const float* Arow = X + (size_t)(tm * 16 + l16) * HDIM + 2 * kh;
  const float* Bcol = Wcat + (size_t)(2 * kh) * NOUT + tn * 16 + l16;

  v8f acc = {};
#pragma unroll
  for (int k0 = 0; k0 < HDIM; k0 += 4) {
    v2f a = *(const v2f*)(Arow + k0);           // K = k0+2*kh, k0+2*kh+1
    v2f b;
    b.x = Bcol[(size_t)k0 * NOUT];              // row k0+2*kh
    b.y = Bcol[(size_t)k0 * NOUT + NOUT];       // row k0+2*kh+1
    acc = __builtin_amdgcn_wmma_f32_16x16x4_f32(
        false, a, false, b, (short)0, acc, false, false);
  }

  int n = tn * 16 + l16;
  float e0 = E0[n], e1 = E1[n], bb = Bias[n];
#pragma unroll
  for (int r = 0; r < 8; ++r) {
    int m = tm * 16 + r + 8 * kh;               // C/D layout: M = r + 8*(lane/16)
    float c0 = centers[m * 2 + 0];
    float c1 = centers[m * 2 + 1];
    G[(size_t)m * NOUT + n] = acc[r] + c0 * e0 + c1 * e1 + bb;
  }
}

// ---------------------------------------------------------------------------
// Message + aggregate: one block per sample (16 agents). For target i, chan h:
//   agg[i][h] = sum_{j!=i} sigmoid(Pf[i][h]+Qf[j][h]) * softplus(Ps[i][h]+Qs[j][h])
// Q rows cached in LDS; per-channel sum/sumsq reduced and atomically added to
// global BN stats.
// ---------------------------------------------------------------------------
__global__ __launch_bounds__(256) void msg_agg_kernel(
    const float* __restrict__ G, float* __restrict__ agg,
    float* __restrict__ stats) {
  __shared__ float qf[AGENTS * HDIM];
  __shared__ float qs[AGENTS * HDIM];
  __shared__ float red[512];

  int tid = threadIdx.x;
  int base = blockIdx.x * AGENTS;

#pragma unroll
  for (int v = 0; v < 8; ++v) {
    int idx = tid + v * 256;                    // idx = node_local*128 + h
    int nl = idx >> 7;
    int hh = idx & 127;
    qf[idx] = G[(size_t)(base + nl) * NOUT + 128 + hh];
    qs[idx] = G[(size_t)(base + nl) * NOUT + 384 + hh];
  }
  __syncthreads();

  int h = tid & 127;
  int ig = tid >> 7;                            // handles i = ig*8 .. ig*8+7
  float pf[8], ps[8], acc[8];
#pragma unroll
  for (int r = 0; r < 8; ++r) {
    int i = ig * 8 + r;
    pf[r] = G[(size_t)(base + i) * NOUT + h];
    ps[r] = G[(size_t)(base + i) * NOUT + 256 + h];
    acc[r] = 0.0f;
  }

  for (int j = 0; j < AGENTS; ++j) {
    float qfv = qf[j * HDIM + h];
    float qsv = qs[j * HDIM + h];
#pragma unroll
    for (int r = 0; r < 8; ++r) {
      int i = ig * 8 + r;
      float f = pf[r] + qfv;
      float s = ps[r] + qsv;
      float sig = 1.0f / (1.0f + __expf(-f));
      float sp = fmaxf(s, 0.0f) + __logf(1.0f + __expf(-fabsf(s)));
      acc[r] += (j == i) ? 0.0f : sig * sp;
    }
  }

  float lsum = 0.0f, lsq = 0.0f;
#pragma unroll
  for (int r = 0; r < 8; ++r) {
    int i = ig * 8 + r;
    agg[(size_t)(base + i) * HDIM + h] = acc[r];
    lsum += acc[r];
    lsq += acc[r] * acc[r];
  }
  red[tid] = lsum;
  red[256 + tid] = lsq;
  __syncthreads();
  if (tid < 128) {
    atomicAdd(&stats[tid], red[tid] + red[tid + 128]);
    atomicAdd(&stats[128 + tid], red[256 + tid] + red[384 + tid]);
  }
}

// ---------------------------------------------------------------------------
// BatchNorm (train-mode batch stats) + residual + ReLU, fully elementwise.
// ---------------------------------------------------------------------------
__global__ __launch_bounds__(256) void bn_relu_kernel(
    const float* __restrict__ agg, const float* __restrict__ x,
    const float* __restrict__ stats, const float* __restrict__ gamma,
    const float* __restrict__ beta, float* __restrict__ out) {
  int idx = blockIdx.x * 256 + threadIdx.x;     // N*H = 2097152
  int h = idx & 127;
  const float invN = 1.0f / (float)NNODES;
  float mean = stats[h] * invN;
  float var = stats[128 + h] * invN - mean * mean;
  float scl = gamma[h] * rsqrtf(var + BN_EPS);
  float v = (agg[idx] - mean) * scl + beta[h] + x[idx];
  out[idx] = fmaxf(v, 0.0f);
}

// ---------------------------------------------------------------------------
extern "C" void kernel_launch(void* const* d_in, const int* in_sizes, int n_in,
                              void* d_out, int out_size, void* d_ws, size_t ws_size,
                              hipStream_t stream) {
  (void)in_sizes; (void)n_in; (void)out_size; (void)ws_size;
  const float* x0      = (const float*)d_in[0];   // gnn_in [N,128]
  const float* centers = (const float*)d_in[1];   // [N,2]
  // d_in[2] = edge_index: structure is known (dense per sample), not needed
  const float* wf1 = (const float*)d_in[3];
  const float* bf1 = (const float*)d_in[4];
  const float* ws1 = (const float*)d_in[5];
  const float* bs1 = (const float*)d_in[6];
  const float* g1  = (const float*)d_in[7];
  const float* be1 = (const float*)d_in[8];
  const float* wf2 = (const float*)d_in[9];
  const float* bf2 = (const float*)d_in[10];
  const float* ws2 = (const float*)d_in[11];
  const float* bs2 = (const float*)d_in[12];
  const float* g2  = (const float*)d_in[13];
  const float* be2 = (const float*)d_in[14];
  float* out = (float*)d_out;

  float* wsf   = (float*)d_ws;
  float* Wcat  = wsf;                       // 65536 floats
  float* E0    = wsf + 65536;               // 512
  float* E1    = wsf + 66048;               // 512
  float* Bias  = wsf + 66560;               // 512
  float* stats = wsf + 67072;               // 256
  float* G     = wsf + 131072;              // N*512 = 8388608
  float* agg   = G + (size_t)NNODES * NOUT; // N*128 = 2097152
  float* x1    = agg + (size_t)NNODES * HDIM;

  const int gemm_blocks = (NNODES / 16) * (NOUT / 16) / 8;  // 4096
  const int ew_blocks = NNODES * HDIM / 256;                // 8192

  // ---- layer 1 ----
  pack_kernel<<<256, 256, 0, stream>>>(wf1, bf1, ws1, bs1, Wcat, E0, E1, Bias, stats);
  gemm_pq_kernel<<<gemm_blocks, 256, 0, stream>>>(x0, centers, Wcat, E0, E1, Bias, G);
  msg_agg_kernel<<<NSAMP, 256, 0, stream>>>(G, agg, stats);
  bn_relu_kernel<<<ew_blocks, 256, 0, stream>>>(agg, x0, stats, g1, be1, x1);

  // ---- layer 2 ----
  pack_kernel<<<256, 256, 0, stream>>>(wf2, bf2, ws2, bs2, Wcat, E0, E1, Bias, stats);
  gemm_pq_kernel<<<gemm_blocks, 256, 0, stream>>>(x1, centers, Wcat, E0, E1, Bias, G);
  msg_agg_kernel<<<NSAMP, 256, 0, stream>>>(G, agg, stats);
  bn_relu_kernel<<<ew_blocks, 256, 0, stream>>>(agg, x1, stats, g2, be2, out);
}